// MultiheadedSelfAttentionLayer_10411000726285
// MI455X (gfx1250) — compile-verified
//
#include <hip/hip_runtime.h>
#include <math.h>

#define V_NODES 50000
#define N_EDGES 800000

typedef __attribute__((ext_vector_type(2))) float v2f;
typedef __attribute__((ext_vector_type(8))) float v8f;

// -------------------------------------------------------------------------
// GEMM: out[M x 128] = A[M x 128] @ W[128 x 128] + bias[128]
// 4 waves / block; each wave computes one 16x16 tile with
// v_wmma_f32_16x16x4_f32 (32 K-steps). Full fp32 precision.
// -------------------------------------------------------------------------
__global__ __launch_bounds__(128)
void gemm128_wmma(const float* __restrict__ A, const float* __restrict__ W,
                  const float* __restrict__ bias, float* __restrict__ out, int M) {
  const int wave = threadIdx.x >> 5;
  const int lane = threadIdx.x & 31;
  const int half = lane >> 4;          // 0 or 1
  const int m    = lane & 15;
  const int col0 = blockIdx.y * 64 + wave * 16;
  const int rowBase = blockIdx.x * 16;

  int row = rowBase + m;
  if (row >= M) row = M - 1;           // clamp: keep EXEC all-ones for WMMA

  v8f acc = {};
  const float* arow = A + (long)row * 128 + 2 * half;       // A[m][k+2*half .. +1]
  const float* wcol = W + (long)(2 * half) * 128 + col0 + m;

  #pragma unroll
  for (int k = 0; k < 128; k += 4) {
    v2f a = *(const v2f*)(arow + k);   // K = k+2*half, k+2*half+1
    v2f b;
    b.x = wcol[(long)k * 128];         // W[k+2*half][n]
    b.y = wcol[(long)(k + 1) * 128];   // W[k+2*half+1][n]
    acc = __builtin_amdgcn_wmma_f32_16x16x4_f32(
        /*neg_a=*/false, a, /*neg_b=*/false, b,
        /*c_mod=*/(short)0, acc, /*reuse_a=*/false, /*reuse_b=*/false);
  }

  const float bb = bias[col0 + m];
  #pragma unroll
  for (int r = 0; r < 8; ++r) {
    int orow = rowBase + r + 8 * half; // C/D layout: M = r + 8*(lane/16)
    if (orow < M) out[(long)orow * 128 + col0 + m] = acc[r] + bb;
  }
}

// -------------------------------------------------------------------------
// Init: Hacc = 0, m = -inf, denom = 0
// -------------------------------------------------------------------------
__global__ __launch_bounds__(256)
void init_kernel(float* __restrict__ Hacc, float* __restrict__ mbuf,
                 float* __restrict__ denom) {
  long i = (long)blockIdx.x * blockDim.x + threadIdx.x;
  if (i < (long)V_NODES * 128) Hacc[i] = 0.0f;
  if (i < (long)V_NODES * 8) { mbuf[i] = -INFINITY; denom[i] = 0.0f; }
}

// -------------------------------------------------------------------------
// Edge bias: bias[e] = dot(edge_feats[e,:64], W_b) + b_b  (16 lanes / edge)
// -------------------------------------------------------------------------
__global__ __launch_bounds__(256)
void bias_kernel(const float* __restrict__ ef, const float* __restrict__ Wb,
                 const float* __restrict__ bb, float* __restrict__ bias, int E) {
  long t = (long)blockIdx.x * blockDim.x + threadIdx.x;
  int e = (int)(t >> 4);
  int g = (int)(t & 15);
  if (e >= E) return;
  const float4 x = ((const float4*)(ef + (long)e * 64))[g];
  const float4 w = ((const float4*)Wb)[g];
  float p = x.x * w.x + x.y * w.y + x.z * w.z + x.w * w.w;
  p += __shfl_xor(p, 1, 32);
  p += __shfl_xor(p, 2, 32);
  p += __shfl_xor(p, 4, 32);
  p += __shfl_xor(p, 8, 32);
  if (g == 0) bias[e] = p + bb[0];
}

// -------------------------------------------------------------------------
// Float atomic max via signed-max / unsigned-min bit trick (IEEE ordering).
// -------------------------------------------------------------------------
__device__ __forceinline__ void atomicMaxF(float* addr, float val) {
  if (val >= 0.0f)
    atomicMax((int*)addr, __float_as_int(val));
  else
    atomicMin((unsigned int*)addr, __float_as_uint(val));
}

// -------------------------------------------------------------------------
// Scores: one wave per edge. score[e][h] = dot(Q[src],K[dst])_h / 8 + bias[e]
// then segment-max into mbuf[dst][h].
// -------------------------------------------------------------------------
__global__ __launch_bounds__(256)
void scores_kernel(const float* __restrict__ Qn, const float* __restrict__ Kn,
                   const float* __restrict__ bias, const int* __restrict__ src,
                   const int* __restrict__ dst, float* __restrict__ scores,
                   float* __restrict__ mbuf, int E) {
  int wid  = (int)(((long)blockIdx.x * blockDim.x + threadIdx.x) >> 5);
  int lane = threadIdx.x & 31;
  if (wid >= E) return;
  int s = src[wid], d = dst[wid];
  float4 q = ((const float4*)(Qn + (long)s * 128))[lane];
  float4 k = ((const float4*)(Kn + (long)d * 128))[lane];
  float p = q.x * k.x + q.y * k.y + q.z * k.z + q.w * k.w;
  p += __shfl_xor(p, 1, 32);   // reduce 4 lanes -> one head (16 dims)
  p += __shfl_xor(p, 2, 32);
  if ((lane & 3) == 0) {
    int h = lane >> 2;
    float sc = p * 0.125f + bias[wid];   // SQRT_DK == 8.0 (faithful)
    scores[(long)wid * 8 + h] = sc;
    atomicMaxF(&mbuf[(long)d * 8 + h], sc);
  }
}

// -------------------------------------------------------------------------
// e = exp(score - m[dst]); denom[dst] += e  (scores overwritten with e)
// -------------------------------------------------------------------------
__global__ __launch_bounds__(256)
void exp_kernel(const int* __restrict__ dst, float* __restrict__ scores,
                const float* __restrict__ mbuf, float* __restrict__ denom, int E) {
  long t = (long)blockIdx.x * blockDim.x + threadIdx.x;
  if (t >= (long)E * 8) return;
  int e = (int)(t >> 3), h = (int)(t & 7);
  int d = dst[e];
  float ev = __expf(scores[t] - mbuf[(long)d * 8 + h]);
  scores[t] = ev;
  atomicAdd(&denom[(long)d * 8 + h], ev);
}

// -------------------------------------------------------------------------
// Hacc[dst][c] += (e/denom[dst][h]) * Vn[dst][c]   (c = h*16 + d16)
// -------------------------------------------------------------------------
__global__ __launch_bounds__(256)
void scatter_kernel(const int* __restrict__ dst, const float* __restrict__ scores,
                    const float* __restrict__ denom, const float* __restrict__ Vn,
                    float* __restrict__ Hacc, int E) {
  long t = (long)blockIdx.x * blockDim.x + threadIdx.x;
  if (t >= (long)E * 128) return;
  int e = (int)(t >> 7), c = (int)(t & 127), h = c >> 4;
  int d = dst[e];
  float alpha = scores[(long)e * 8 + h] / denom[(long)d * 8 + h];
  atomicAdd(&Hacc[(long)d * 128 + c], alpha * Vn[(long)d * 128 + c]);
}

// -------------------------------------------------------------------------
extern "C" void kernel_launch(void* const* d_in, const int* in_sizes, int n_in,
                              void* d_out, int out_size, void* d_ws, size_t ws_size,
                              hipStream_t stream) {
  const float* node_feats = (const float*)d_in[0];
  const float* edge_feats = (const float*)d_in[1];
  const int*   edge_index = (const int*)d_in[2];
  const float* W_q = (const float*)d_in[3];
  const float* b_q = (const float*)d_in[4];
  const float* W_k = (const float*)d_in[5];
  const float* b_k = (const float*)d_in[6];
  const float* W_v = (const float*)d_in[7];
  const float* b_v = (const float*)d_in[8];
  const float* W_b = (const float*)d_in[9];
  const float* b_b = (const float*)d_in[10];
  const float* W_o = (const float*)d_in[11];
  const float* b_o = (const float*)d_in[12];
  float* out = (float*)d_out;

  const int V = V_NODES, E = N_EDGES;
  const int* src = edge_index;
  const int* dst = edge_index + E;

  // workspace layout (floats): Qn, Kn, Vn, Hacc [V*128 each], m, denom [V*8],
  // bias [E], scores [E*8]  => ~134 MB
  float* ws    = (float*)d_ws;
  float* Qn    = ws;
  float* Kn    = Qn + (long)V * 128;
  float* Vn    = Kn + (long)V * 128;
  float* Hacc  = Vn + (long)V * 128;
  float* mbuf  = Hacc + (long)V * 128;
  float* denom = mbuf + (long)V * 8;
  float* biasE = denom + (long)V * 8;
  float* scores = biasE + (long)E;

  {
    long n = (long)V * 128;
    init_kernel<<<(int)((n + 255) / 256), 256, 0, stream>>>(Hacc, mbuf, denom);
  }

  dim3 gGrid((V + 15) / 16, 2), gBlk(128);
  gemm128_wmma<<<gGrid, gBlk, 0, stream>>>(node_feats, W_q, b_q, Qn, V);
  gemm128_wmma<<<gGrid, gBlk, 0, stream>>>(node_feats, W_k, b_k, Kn, V);
  gemm128_wmma<<<gGrid, gBlk, 0, stream>>>(node_feats, W_v, b_v, Vn, V);

  bias_kernel<<<(int)(((long)E * 16 + 255) / 256), 256, 0, stream>>>(
      edge_feats, W_b, b_b, biasE, E);
  scores_kernel<<<(int)(((long)E * 32 + 255) / 256), 256, 0, stream>>>(
      Qn, Kn, biasE, src, dst, scores, mbuf, E);
  exp_kernel<<<(int)(((long)E * 8 + 255) / 256), 256, 0, stream>>>(
      dst, scores, mbuf, denom, E);
  scatter_kernel<<<(int)(((long)E * 128 + 255) / 256), 256, 0, stream>>>(
      dst, scores, denom, Vn, Hacc, E);

  gemm128_wmma<<<gGrid, gBlk, 0, stream>>>(Hacc, W_o, b_o, out, V);
}